// SingleHeadAttention_58351425683698
// MI455X (gfx1250) — compile-verified
//
#include <hip/hip_runtime.h>
#include <hip/hip_bf16.h>
#include <math.h>

// Single-head causal attention for B=4, S=4096, E=1024, A=128 (all fp32).
// Strategy: full-precision CDNA5 matrix path via V_WMMA_F32_16X16X4_F32.
//  Pass 1: Q/K/V projections (WMMA GEMM) into workspace (3 x 8 MB).
//  Pass 2: flash-attention (online softmax), 1 wave per 16-row query tile.

typedef float v2f __attribute__((ext_vector_type(2)));
typedef float v8f __attribute__((ext_vector_type(8)));

#define EMBED 1024
#define SEQ   4096
#define ADIM  128
#define BATCH 4

static __device__ inline v8f wmma_f32x4(v2f a, v2f b, v8f c) {
    // D = A(16x4 f32) * B(4x16 f32) + C(16x16 f32)
    return __builtin_amdgcn_wmma_f32_16x16x4_f32(
        /*neg_a=*/false, a, /*neg_b=*/false, b,
        /*c_mod=*/(short)0, c, /*reuse_a=*/false, /*reuse_b=*/false);
}

// ---------------------------------------------------------------------------
// Pass 1: O[m, 0:128] = X[m, 0:1024] @ W[0:1024, 0:128]
// grid.x: row tile (16 rows each) over B*S = 16384 rows -> 1024
// grid.y: which matrix (0=Q uses Wq, 1=K uses Wk, 2=V uses Wv)
// block: 32 threads (1 wave)
// ---------------------------------------------------------------------------
__global__ void __launch_bounds__(32)
proj_kernel(const float* __restrict__ X,
            const float* __restrict__ Wq,
            const float* __restrict__ Wk,
            const float* __restrict__ Wv,
            float* __restrict__ Q,
            float* __restrict__ K,
            float* __restrict__ V) {
    const int lane = threadIdx.x;
    const int r = lane & 15;      // row within tile (A) / col (B/C)
    const int h = lane >> 4;      // lane-half selector
    const int m0 = blockIdx.x * 16;
    const int mat = blockIdx.y;

    const float* W = (mat == 0) ? Wq : (mat == 1) ? Wk : Wv;
    float*       O = (mat == 0) ? Q  : (mat == 1) ? K  : V;

    const v8f zero = {0.f, 0.f, 0.f, 0.f, 0.f, 0.f, 0.f, 0.f};
    v8f acc[8];
#pragma unroll
    for (int nt = 0; nt < 8; ++nt) acc[nt] = zero;

    const float* xrow = X + (size_t)(m0 + r) * EMBED;

    for (int k0 = 0; k0 < EMBED; k0 += 4) {
        const int ka = k0 + 2 * h;
        v2f a = *(const v2f*)(xrow + ka);                 // A[r][ka], A[r][ka+1]
        const float* w0 = W + (size_t)ka * ADIM;          // row ka of W
        const float* w1 = w0 + ADIM;                      // row ka+1
#pragma unroll
        for (int nt = 0; nt < 8; ++nt) {
            v2f b;
            b.x = w0[nt * 16 + r];
            b.y = w1[nt * 16 + r];
            acc[nt] = wmma_f32x4(a, b, acc[nt]);
        }
    }

#pragma unroll
    for (int nt = 0; nt < 8; ++nt)
#pragma unroll
        for (int j = 0; j < 8; ++j)
            O[(size_t)(m0 + j + 8 * h) * ADIM + nt * 16 + r] = acc[nt][j];
}

// ---------------------------------------------------------------------------
// Pass 2: flash attention, causal. grid = (S/16, B), block = 32 (1 wave).
// Each wave owns one 16x128 output tile; iterates key tiles t = 0..qt.
// ---------------------------------------------------------------------------
__global__ void __launch_bounds__(32)
attn_kernel(const float* __restrict__ Q,
            const float* __restrict__ K,
            const float* __restrict__ V,
            float* __restrict__ out) {
    const int lane = threadIdx.x;
    const int c = lane & 15;
    const int h = lane >> 4;
    const int qt = blockIdx.x;               // query tile
    const int bb = blockIdx.y;               // batch
    const size_t base = (size_t)bb * SEQ * ADIM;
    const int q0 = qt * 16;

    __shared__ float pS[16 * 16];            // P tile staging for transpose

    // Preload Q fragments for all 32 K-steps of the 128-deep dot product.
    v2f qf[32];
#pragma unroll
    for (int kk = 0; kk < 32; ++kk)
        qf[kk] = *(const v2f*)(Q + base + (size_t)(q0 + c) * ADIM + kk * 4 + 2 * h);

    const v8f zero = {0.f, 0.f, 0.f, 0.f, 0.f, 0.f, 0.f, 0.f};
    v8f o[8];
#pragma unroll
    for (int nt = 0; nt < 8; ++nt) o[nt] = zero;

    float m[8], l[8];
#pragma unroll
    for (int j = 0; j < 8; ++j) { m[j] = -INFINITY; l[j] = 0.f; }

    const float scale = 0.08838834764831844f; // 1/sqrt(128)

    for (int t = 0; t <= qt; ++t) {
        // ---- S tile = Q_tile @ K_tile^T  (16x16, f32 accum) ----
        v8f s = zero;
        const float* Kt = K + base + (size_t)t * 16 * ADIM;
#pragma unroll
        for (int kk = 0; kk < 32; ++kk) {
            // B[kd][n] = K[t*16+n][kd] -> contiguous float2 in row-major K
            v2f bfr = *(const v2f*)(Kt + (size_t)c * ADIM + kk * 4 + 2 * h);
            s = wmma_f32x4(qf[kk], bfr, s);
        }

        // ---- scale + causal mask (diagonal tile only) ----
        float p[8];
#pragma unroll
        for (int j = 0; j < 8; ++j) {
            float sv = s[j] * scale;
            if (t == qt && c > j + 8 * h) sv = -INFINITY; // key col > query row
            p[j] = sv;
        }

        // ---- online softmax per row (row lives across a 16-lane half) ----
#pragma unroll
        for (int j = 0; j < 8; ++j) {
            float v = p[j];
            v = fmaxf(v, __shfl_xor(v, 1, 32));
            v = fmaxf(v, __shfl_xor(v, 2, 32));
            v = fmaxf(v, __shfl_xor(v, 4, 32));
            v = fmaxf(v, __shfl_xor(v, 8, 32));
            const float mnew  = fmaxf(m[j], v);
            const float alpha = __expf(m[j] - mnew);
            const float pe    = __expf(p[j] - mnew);
            float rs = pe;
            rs += __shfl_xor(rs, 1, 32);
            rs += __shfl_xor(rs, 2, 32);
            rs += __shfl_xor(rs, 4, 32);
            rs += __shfl_xor(rs, 8, 32);
            l[j] = l[j] * alpha + rs;
            m[j] = mnew;
            p[j] = pe;
#pragma unroll
            for (int nt = 0; nt < 8; ++nt) o[nt][j] *= alpha;
        }

        // ---- transpose P (D-layout -> A-layout) through LDS ----
        __syncthreads();
#pragma unroll
        for (int j = 0; j < 8; ++j)
            pS[(j + 8 * h) * 16 + c] = p[j];
        __syncthreads();
        v2f pf[4];
#pragma unroll
        for (int kk = 0; kk < 4; ++kk) {
            pf[kk].x = pS[c * 16 + kk * 4 + 2 * h];
            pf[kk].y = pS[c * 16 + kk * 4 + 2 * h + 1];
        }

        // ---- O += P @ V_tile ----
        const float* Vt = V + base + (size_t)t * 16 * ADIM;
#pragma unroll
        for (int nt = 0; nt < 8; ++nt) {
#pragma unroll
            for (int kk = 0; kk < 4; ++kk) {
                v2f bv;
                bv.x = Vt[(size_t)(kk * 4 + 2 * h)     * ADIM + nt * 16 + c];
                bv.y = Vt[(size_t)(kk * 4 + 2 * h + 1) * ADIM + nt * 16 + c];
                o[nt] = wmma_f32x4(pf[kk], bv, o[nt]);
            }
        }
    }

    // ---- epilogue: normalize, round to 4 decimals like reference ----
#pragma unroll
    for (int nt = 0; nt < 8; ++nt)
#pragma unroll
        for (int j = 0; j < 8; ++j) {
            float val = o[nt][j] / l[j];
            val = rintf(val * 1.0e4f) * 1.0e-4f;
            out[base + (size_t)(q0 + j + 8 * h) * ADIM + nt * 16 + c] = val;
        }
}

// ---------------------------------------------------------------------------
extern "C" void kernel_launch(void* const* d_in, const int* in_sizes, int n_in,
                              void* d_out, int out_size, void* d_ws, size_t ws_size,
                              hipStream_t stream) {
    // setup_inputs order: embedded, Wk, Wq, Wv
    const float* X  = (const float*)d_in[0];
    const float* Wk = (const float*)d_in[1];
    const float* Wq = (const float*)d_in[2];
    const float* Wv = (const float*)d_in[3];

    float* ws = (float*)d_ws;
    const size_t per = (size_t)BATCH * SEQ * ADIM;   // 2M floats = 8 MB
    float* Qb = ws;
    float* Kb = ws + per;
    float* Vb = ws + 2 * per;                        // needs 24 MB of workspace

    dim3 g1((BATCH * SEQ) / 16, 3);
    proj_kernel<<<g1, 32, 0, stream>>>(X, Wq, Wk, Wv, Qb, Kb, Vb);

    dim3 g2(SEQ / 16, BATCH);
    attn_kernel<<<g2, 32, 0, stream>>>(Qb, Kb, Vb, (float*)d_out);
}